// PoolingModule_86114094285202
// MI455X (gfx1250) — compile-verified
//
#include <hip/hip_runtime.h>
#include <math.h>
#include <limits.h>

// Token-merge (ToMe-style) sequential pooling for MI455X / gfx1250.
// One workgroup per batch sample; dist/index arrays live in LDS; row data in
// d_ws (L2-resident). Initial adjacent-pair dots are batched on the matrix
// pipe via V_WMMA_F32_16X16X4_F32 (diagonal of X[j:j+16] @ X[j+1:j+17]^T).

#define NT     512          // threads per block = 16 wave32s
#define SEQ    1024
#define DIM    512
#define NCOMP  256
#define STEPS  (SEQ - NCOMP)   // 768 sequential merge steps

typedef __attribute__((ext_vector_type(2))) float v2f;
typedef __attribute__((ext_vector_type(8))) float v8f;

__device__ __forceinline__ void argmax_combine(float& v, int& ix, float vo, int io) {
    // max value, first (smallest) index on ties -> matches jnp.argmax
    if (vo > v || (vo == v && io < ix)) { v = vo; ix = io; }
}

__global__ __launch_bounds__(NT) void tome_merge_kernel(const float* __restrict__ xin,
                                                        float* __restrict__ out,
                                                        float* __restrict__ ws) {
    const int b    = blockIdx.x;
    const int tid  = threadIdx.x;
    const int lane = tid & 31;
    const int wave = tid >> 5;          // 0..15

    float*       xb = ws  + (size_t)b * SEQ * DIM;   // mutable row storage
    const float* xi = xin + (size_t)b * SEQ * DIM;

    __shared__ float dist[SEQ];        // adjacent-pair dot products (logical order)
    __shared__ int   idxp[SEQ];        // logical position -> physical row
    __shared__ float mrow[DIM];        // staged merged row
    __shared__ float tile[16][16 * 16];// per-wave 16x16 WMMA accumulator tile
    __shared__ float wval[16];
    __shared__ int   widx[16];
    __shared__ float wsum[16];
    __shared__ int   smid;

    // ---- copy this batch into the workspace (inputs must stay unmodified) ----
    for (int i = tid; i < SEQ * DIM; i += NT) xb[i] = xi[i];
    for (int i = tid; i < SEQ; i += NT) idxp[i] = i;
    __threadfence_block();
    __syncthreads();

    // ---- initial dist[j] = dot(x[j], x[j+1]) via V_WMMA_F32_16X16X4_F32 ----
    // 64 groups of 16 pairs; each of the 16 waves handles 4 groups.
    for (int rep = 0; rep < 4; ++rep) {
        const int g  = wave + rep * 16;          // 0..63
        const int j0 = g * 16;
        const int M  = lane & 15;
        const int ko = (lane < 16) ? 0 : 2;      // A/B K-pair split across half-waves
        const int rb_row = (j0 + M + 1 < SEQ) ? (j0 + M + 1) : (SEQ - 1);
        const float* ra = xb + (size_t)(j0 + M) * DIM + ko;
        const float* rbp = xb + (size_t)rb_row   * DIM + ko;

        v8f c = {};
        for (int k = 0; k < DIM; k += 4) {
            v2f a, bb;
            a.x  = ra[k];  a.y  = ra[k + 1];     // A: 16x4, lanes 0-15 K0/K1, 16-31 K2/K3
            bb.x = rbp[k]; bb.y = rbp[k + 1];    // B: 4x16, col striped across lanes
            c = __builtin_amdgcn_wmma_f32_16x16x4_f32(
                    false, a, false, bb, (short)0, c, false, false);
        }
        // Scatter C tile: VGPR gg holds row M = gg + (lane<16 ? 0 : 8), col = lane&15
        float* tw = tile[wave];
        const int rbase = (lane < 16) ? 0 : 8;
#pragma unroll
        for (int gg = 0; gg < 8; ++gg)
            tw[(rbase + gg) * 16 + M] = c[gg];
        __syncthreads();                         // order tile store -> diagonal read
        if (lane < 16) {
            const int j = j0 + lane;
            if (j < SEQ - 1) dist[j] = tw[lane * 16 + lane];
        }
        __syncthreads();                         // protect tile reuse next rep
    }

    // ---- 768 sequential merge steps ----
    int len = SEQ;
    for (int t = 0; t < STEPS; ++t) {
        const int npairs = len - 1;

        // (1) argmax over dist[0..npairs-1], first occurrence wins
        float v = -INFINITY; int ix = INT_MAX;
        for (int j = tid; j < npairs; j += NT) argmax_combine(v, ix, dist[j], j);
#pragma unroll
        for (int off = 16; off > 0; off >>= 1) {
            float vo = __shfl_xor(v, off, 32);
            int   io = __shfl_xor(ix, off, 32);
            argmax_combine(v, ix, vo, io);
        }
        if (lane == 0) { wval[wave] = v; widx[wave] = ix; }
        __syncthreads();
        if (wave == 0) {
            const int w = lane & 15;
            v = wval[w]; ix = widx[w];
#pragma unroll
            for (int off = 16; off > 0; off >>= 1) {
                float vo = __shfl_xor(v, off, 32);
                int   io = __shfl_xor(ix, off, 32);
                argmax_combine(v, ix, vo, io);
            }
            if (lane == 0) smid = ix;
        }
        __syncthreads();

        const int mid  = smid;
        const int rowA = idxp[mid];
        const int rowB = idxp[mid + 1];
        const int pl   = (mid > 0)       ? idxp[mid - 1] : -1;  // read before shift
        const int pr   = (mid < len - 2) ? idxp[mid + 2] : -1;

        // (2) merge rows: 512 threads x 1 element (D == NT)
        {
            const float m = 0.5f * (xb[(size_t)rowA * DIM + tid] +
                                    xb[(size_t)rowB * DIM + tid]);
            mrow[tid] = m;
            xb[(size_t)rowA * DIM + tid] = m;
        }
        __threadfence_block();   // release merged-row stores (WGP scope)
        __syncthreads();
        __threadfence_block();   // acquire: don't read stale WGP$ lines

        // (3) the two refreshed dots: waves 0-7 -> left pair, waves 8-15 -> right pair
        {
            float part = 0.0f;
            if (tid < 256) {
                if (pl >= 0) {
                    const int e = tid * 2;
                    part = xb[(size_t)pl * DIM + e]     * mrow[e] +
                           xb[(size_t)pl * DIM + e + 1] * mrow[e + 1];
                }
            } else {
                if (pr >= 0) {
                    const int e = (tid - 256) * 2;
                    part = xb[(size_t)pr * DIM + e]     * mrow[e] +
                           xb[(size_t)pr * DIM + e + 1] * mrow[e + 1];
                }
            }
#pragma unroll
            for (int off = 16; off > 0; off >>= 1)
                part += __shfl_xor(part, off, 32);
            if (lane == 0) wsum[wave] = part;
        }

        // shift reads (tails of dist / idxp) into registers
        float sd0 = 0.f, sd1 = 0.f; int si0 = 0, si1 = 0;
        const int p0   = mid + 1 + tid;
        const int p1   = p0 + NT;
        const int dEnd = len - 3;   // newdist[j] = old dist[j+1] for mid < j <= len-3
        const int iEnd = len - 2;   // newidx[j]  = old idxp[j+1] for mid < j <= len-2
        if (p0 <= dEnd) sd0 = dist[p0 + 1];
        if (p1 <= dEnd) sd1 = dist[p1 + 1];
        if (p0 <= iEnd) si0 = idxp[p0 + 1];
        if (p1 <= iEnd) si1 = idxp[p1 + 1];
        __syncthreads();

        // (4) shift writes + refreshed dist entries
        if (p0 <= dEnd) dist[p0] = sd0;
        if (p1 <= dEnd) dist[p1] = sd1;
        if (p0 <= iEnd) idxp[p0] = si0;
        if (p1 <= iEnd) idxp[p1] = si1;
        if (tid == 0) {
            float dl = 0.f, dr = 0.f;
#pragma unroll
            for (int w = 0; w < 8; ++w)  dl += wsum[w];
#pragma unroll
            for (int w = 8; w < 16; ++w) dr += wsum[w];
            if (mid > 0)       dist[mid - 1] = dl;
            if (mid < len - 2) dist[mid]     = dr;
        }
        __syncthreads();
        len -= 1;
    }

    // ---- emit first NCOMP logical rows ----
    __threadfence_block();
    __syncthreads();
    float* ob = out + (size_t)b * NCOMP * DIM;
    for (int r = 0; r < NCOMP; ++r)
        ob[(size_t)r * DIM + tid] = xb[(size_t)idxp[r] * DIM + tid];
}

extern "C" void kernel_launch(void* const* d_in, const int* in_sizes, int n_in,
                              void* d_out, int out_size, void* d_ws, size_t ws_size,
                              hipStream_t stream) {
    const float* x  = (const float*)d_in[0];
    float* out      = (float*)d_out;
    float* ws       = (float*)d_ws;          // needs B*SEQ*DIM*4 = 8 MB scratch
    const int B     = in_sizes[0] / (SEQ * DIM);
    tome_merge_kernel<<<dim3(B), dim3(NT), 0, stream>>>(x, out, ws);
}